// create_72395968741975
// MI455X (gfx1250) — compile-verified
//
#include <hip/hip_runtime.h>
#include <math.h>
#include <stdint.h>

#define WB   256   // batch size == threads per block (8 wave32 waves)
#define DEG  10

typedef __attribute__((ext_vector_type(4))) uint32_t v4u;
typedef __attribute__((ext_vector_type(8))) uint32_t v8u;

// ---------------- CDNA5 async global->LDS (gfx1250, ASYNCcnt-tracked) --------
__device__ __forceinline__ void async_load_b32_to_lds(uint32_t lds_byte_off,
                                                      const void* gaddr) {
  // copies 4 bytes from the global address in vaddr to the LDS offset in vdst; ASYNCcnt
  asm volatile("global_load_async_to_lds_b32 %0, %1, off"
               :: "v"(lds_byte_off), "v"((uint64_t)(uintptr_t)gaddr)
               : "memory");
}
__device__ __forceinline__ void wait_asynccnt0() {
  asm volatile("s_wait_asynccnt 0" ::: "memory");
}
// generic pointer to a __shared__ object: low 32 bits are the LDS byte offset
__device__ __forceinline__ uint32_t lds_off_of(const void* p) {
  return (uint32_t)(uintptr_t)p;
}

// ---------------- CDNA5 Tensor Data Mover (TENSORcnt-tracked) ---------------
// 1-D contiguous tile: nelem f32 from gaddr to the LDS byte offset lds_byte_off.
// D# built per CDNA5 ISA (group0 + group1, VADDR2/3 off).
__device__ __forceinline__ void tdm_load_f32_tile(uint32_t lds_byte_off,
                                                  const void* gaddr,
                                                  uint32_t nelem) {
  const uint64_t ga = (uint64_t)(uintptr_t)gaddr;
  v4u g0;
  g0.x = 1u;                                            // count=1 (valid), user D#
  g0.y = lds_byte_off;                                  // lds_addr
  g0.z = (uint32_t)ga;                                  // global_addr[31:0]
  g0.w = (uint32_t)((ga >> 32) & 0x1FFFFFFu) | (2u << 30); // addr[56:32] | type=2
  v8u g1;
  g1.s0 = 2u << 16;                                     // data_size=4B; mask=0
  g1.s1 = (nelem & 0xFFFFu) << 16;                      // tensor_dim0[15:0]
  g1.s2 = (nelem >> 16) | (1u << 16);                   // tensor_dim0[31:16] | tensor_dim1=1
  g1.s3 = (nelem & 0xFFFFu) << 16;                      // tile_dim0 = nelem
  g1.s4 = 1u;                                           // tile_dim1=1, tile_dim2=0
  g1.s5 = nelem;                                        // tensor_dim0_stride[31:0]
  g1.s6 = 0u;                                           // stride[47:32] | dim1_stride lo
  g1.s7 = 0u;
  asm volatile("tensor_load_to_lds %0, %1" :: "s"(g0), "s"(g1) : "memory");
}

__device__ __forceinline__ void atomic_add_f32(float* p, float v) {
  unsafeAtomicAdd(p, v);   // -> native no-return global_atomic_add_f32
}

// ---------------- setup kernels ---------------------------------------------
__global__ void build_uinit(const float* __restrict__ llr0,
                            float* __restrict__ u, int N) {
  size_t total = (size_t)(N + 1) * WB;
  size_t stride = (size_t)gridDim.x * blockDim.x;
  for (size_t g = (size_t)blockIdx.x * blockDim.x + threadIdx.x; g < total; g += stride) {
    int b    = (int)(g & (WB - 1));   // g = c*WB + b
    size_t c = g >> 8;
    u[g] = (c == (size_t)N) ? INFINITY : llr0[(size_t)b * N + c];
  }
}

__global__ void build_sg(const float* __restrict__ synd,
                         float* __restrict__ sg, int M) {
  size_t total = (size_t)M * WB;
  size_t stride = (size_t)gridDim.x * blockDim.x;
  for (size_t g = (size_t)blockIdx.x * blockDim.x + threadIdx.x; g < total; g += stride) {
    int b    = (int)(g & (WB - 1));   // g = m*WB + b
    size_t m = g >> 8;
    sg[g] = (synd[(size_t)b * M + m] == 0.0f) ? 1.0f : -1.0f;
  }
}

__global__ void fill_f4(float4* __restrict__ p, float v, size_t n4) {
  float4 q = make_float4(v, v, v, v);
  size_t stride = (size_t)gridDim.x * blockDim.x;
  for (size_t g = (size_t)blockIdx.x * blockDim.x + threadIdx.x; g < n4; g += stride)
    p[g] = q;
}

__global__ void copy_f4(const float4* __restrict__ s, float4* __restrict__ d, size_t n4) {
  size_t stride = (size_t)gridDim.x * blockDim.x;
  for (size_t g = (size_t)blockIdx.x * blockDim.x + threadIdx.x; g < n4; g += stride)
    d[g] = s[g];
}

__global__ void init_state(int* __restrict__ ni, int* __restrict__ cv) {
  int b = threadIdx.x; ni[b] = -1; cv[b] = 0;
}
__global__ void reset_ok(int* __restrict__ ok) { ok[threadIdx.x] = 1; }

// ---------------- per-iteration kernels -------------------------------------
// one block == one check row m; lane/thread == batch b (coalesced over B).
// Data movement: TDM DMAs the contiguous 10KB message tile into LDS (wave 0,
// TENSORcnt) while every lane async-copies the 10 gathered l_v segments into
// LDS (ASYNCcnt). Compute starts after both engines drain + block barrier.
__global__ void __launch_bounds__(WB) check_update(
    const float* __restrict__ l_cur, float* __restrict__ l_next,
    float* __restrict__ msg, const int* __restrict__ cols,
    const float* __restrict__ sg, float beta, int N)
{
  __shared__ int   s_cols[16];
  __shared__ float s_lv[DEG * WB];    // async-staged gathered l_v segments
  __shared__ float s_msg[DEG * WB];   // TDM-staged contiguous message tile
  const int m = blockIdx.x;
  const int b = threadIdx.x;
  if (b < DEG) s_cols[b] = cols[m * DEG + b];
  __syncthreads();

  int c[DEG];
#pragma unroll
  for (int d = 0; d < DEG; ++d) c[d] = s_cols[d];

  // wave 0: kick the TDM for this row's message tile (issued once, EXEC-agnostic)
  if (threadIdx.x < 32) {
    tdm_load_f32_tile(lds_off_of(s_msg), msg + (size_t)m * DEG * WB, DEG * WB);
  }

  // all lanes: async-copy the 10 coalesced l_v segments into LDS
  const uint32_t lv_base = lds_off_of(s_lv);
#pragma unroll
  for (int d = 0; d < DEG; ++d)
    async_load_b32_to_lds(lv_base + (uint32_t)((d * WB + b) * 4),
                          &l_cur[(size_t)c[d] * WB + b]);

  wait_asynccnt0();
  if (threadIdx.x < 32) __builtin_amdgcn_s_wait_tensorcnt(0);
  __syncthreads();

  float sgn[DEG], ab[DEG];
  float min0 = INFINITY, min1 = INFINITY, sprod = 1.0f;
#pragma unroll
  for (int d = 0; d < DEG; ++d) {
    float av = s_lv[d * WB + b] - s_msg[d * WB + b];  // a (iter1: u_init[col]-0)
    float s  = (av > 0.0f) ? 1.0f : -1.0f;            // sign(0) -> -1 per reference
    float t  = fabsf(av);                             // dummy col -> +inf, inert
    sgn[d] = s; ab[d] = t; sprod *= s;
    if (t < min0) { min1 = min0; min0 = t; }
    else if (t < min1) { min1 = t; }
  }
  const float q = sg[(size_t)m * WB + b] * sprod;
  const size_t mbase = (size_t)m * DEG * WB + b;
#pragma unroll
  for (int d = 0; d < DEG; ++d) {
    float mres = (ab[d] == min0) ? min1 : min0;
    float bd   = beta * sgn[d] * q * mres;
    if (c[d] >= N) bd = 0.0f;                         // mask dummy edges
    msg[mbase + (size_t)d * WB] = bd;
    if (c[d] < N) atomic_add_f32(&l_next[(size_t)c[d] * WB + b], bd);
  }
}

__global__ void __launch_bounds__(WB) parity_check(
    const float* __restrict__ l_v, const int* __restrict__ cols,
    const float* __restrict__ sg, int* __restrict__ ok, int N)
{
  __shared__ int s_cols[16];
  const int m = blockIdx.x;
  const int b = threadIdx.x;
  if (b < DEG) s_cols[b] = cols[m * DEG + b];
  __syncthreads();
  int par = 0;
#pragma unroll
  for (int d = 0; d < DEG; ++d) {
    int cd = s_cols[d];
    if (cd < N) par ^= (l_v[(size_t)cd * WB + b] <= 0.0f) ? 1 : 0;
    // dummy col: l_v[N] == +inf -> e bit 0, contributes nothing
  }
  int sbit = (sg[(size_t)m * WB + b] < 0.0f) ? 1 : 0;
  if (par != sbit) ok[b] = 0;
}

__global__ void conv_update(const int* __restrict__ ok, int* __restrict__ ni,
                            int* __restrict__ cv, int* __restrict__ newly, int it) {
  int b = threadIdx.x;
  int nw = (ok[b] && ni[b] == -1) ? 1 : 0;
  if (nw) { ni[b] = it; cv[b] = 1; }
  newly[b] = nw;
}

__global__ void final_mark(int* __restrict__ ni, int* __restrict__ newly, int max_iter) {
  int b = threadIdx.x;
  int nw = (ni[b] == -1) ? 1 : 0;
  if (nw) ni[b] = max_iter;                      // conv stays 0
  newly[b] = nw;
}

// snapshot l_v / e_v for freshly-converged batches straight into d_out layout
__global__ void __launch_bounds__(WB) snapshot(
    const float* __restrict__ l_v, const int* __restrict__ newly,
    float* __restrict__ out_e, float* __restrict__ out_l, int N)
{
  __shared__ int f;
  const int b = blockIdx.y;
  if (threadIdx.x == 0) f = newly[b];
  __syncthreads();
  if (!f) return;
  const int c = blockIdx.x * WB + threadIdx.x;
  const float l = l_v[(size_t)c * WB + b];
  out_e[(size_t)b * N + c] = (l <= 0.0f) ? 1.0f : 0.0f;
  out_l[(size_t)b * N + c] = l;
}

__global__ void write_misc(const int* __restrict__ ni, const int* __restrict__ cv,
                           float* __restrict__ out_ni, float* __restrict__ out_cv) {
  int b = threadIdx.x;
  out_ni[b] = (float)ni[b];
  out_cv[b] = (float)cv[b];
}

// ---------------- launch ----------------------------------------------------
extern "C" void kernel_launch(void* const* d_in, const int* in_sizes, int n_in,
                              void* d_out, int out_size, void* d_ws, size_t ws_size,
                              hipStream_t stream)
{
  const float* synd    = (const float*)d_in[0];
  const float* llr0    = (const float*)d_in[1];
  const int*   V_c_col = (const int*)d_in[3];
  const int M = in_sizes[2] / DEG;         // 8192
  const int B = in_sizes[0] / M;           // 256 (== WB)
  const int N = in_sizes[1] / B;           // 16384
  const int MAX_ITER = 30;
  (void)n_in; (void)out_size; (void)ws_size;

  // carve workspace (256B-aligned): whole state ~143MB -> L2-resident (192MB)
  char* ws = (char*)d_ws;
  size_t off = 0;
  auto alloc = [&](size_t bytes) -> void* {
    void* p = ws + off; off += (bytes + 255) & ~(size_t)255; return p;
  };
  const size_t nuB = (size_t)(N + 1) * WB;
  float* u_init = (float*)alloc(nuB * 4);
  float* bufA   = (float*)alloc(nuB * 4);
  float* bufB   = (float*)alloc(nuB * 4);
  float* msg    = (float*)alloc((size_t)M * DEG * WB * 4);
  float* sg     = (float*)alloc((size_t)M * WB * 4);
  int*   ok     = (int*)alloc(WB * 4);
  int*   newly  = (int*)alloc(WB * 4);
  int*   ni     = (int*)alloc(WB * 4);
  int*   cv     = (int*)alloc(WB * 4);

  float* out_e  = (float*)d_out;                       // (B,N)
  float* out_ni = out_e + (size_t)WB * N;              // (B,)
  float* out_l  = out_ni + WB;                         // (B,N)
  float* out_cv = out_l + (size_t)WB * N;              // (B,)

  build_uinit<<<2048, WB, 0, stream>>>(llr0, u_init, N);
  build_sg  <<<2048, WB, 0, stream>>>(synd, sg, M);
  fill_f4   <<<2048, WB, 0, stream>>>((float4*)msg, 0.0f, (size_t)M * DEG * WB / 4);
  init_state<<<1, WB, 0, stream>>>(ni, cv);

  for (int it = 1; it <= MAX_ITER; ++it) {
    const float beta = 1.0f - exp2f(-(float)it);
    const float* l_cur = (it == 1) ? u_init : ((it & 1) ? bufB : bufA);
    float* l_next      = (it & 1) ? bufA : bufB;
    copy_f4     <<<2048, WB, 0, stream>>>((const float4*)u_init, (float4*)l_next, nuB / 4);
    check_update<<<M, WB, 0, stream>>>(l_cur, l_next, msg, V_c_col, sg, beta, N);
    reset_ok    <<<1, WB, 0, stream>>>(ok);
    parity_check<<<M, WB, 0, stream>>>(l_next, V_c_col, sg, ok, N);
    conv_update <<<1, WB, 0, stream>>>(ok, ni, cv, newly, it);
    snapshot    <<<dim3(N / WB, WB), WB, 0, stream>>>(l_next, newly, out_e, out_l, N);
  }
  float* l_final = (MAX_ITER & 1) ? bufA : bufB;
  final_mark<<<1, WB, 0, stream>>>(ni, newly, MAX_ITER);
  snapshot  <<<dim3(N / WB, WB), WB, 0, stream>>>(l_final, newly, out_e, out_l, N);
  write_misc<<<1, WB, 0, stream>>>(ni, cv, out_ni, out_cv);
}